// MultiTaskLossWithMask_81741817577830
// MI455X (gfx1250) — compile-verified
//
#include <hip/hip_runtime.h>
#include <hip/hip_bf16.h>
#include <math.h>

// ---------------------------------------------------------------------------
// MultiTaskLossWithMask for MI455X (gfx1250, wave32).
// Bandwidth-bound multi-reduction: ~140MB effective traffic -> ~6-10us at
// 23.3 TB/s. Wave reductions use V_WMMA_F32_16X16X4_F32 (ones-matrix trick)
// so the reduce runs on the matrix pipe in full f32 precision.
// ---------------------------------------------------------------------------

typedef float v2f __attribute__((ext_vector_type(2)));
typedef float v8f __attribute__((ext_vector_type(8)));

#define HW     (512 * 512)
#define NBATCH 8
#define NC     13
#define NL     12
#define TSIG   5000

// workspace float offsets
#define WS_CE_SUM 0
#define WS_CE_CNT 1
#define WS_INTER  2                    // 104 = 8*13
#define WS_SUMP   (WS_INTER + 104)
#define WS_CNTOH  (WS_SUMP + 104)
#define WS_GRID   (WS_CNTOH + 104)    // 4: bce, inter, sum_p, sum_t
#define WS_BASE   (WS_GRID + 4)       // 1 (+1 pad)
#define WS_SIG    (WS_BASE + 2)       // 96 pairs * 8 floats
#define WS_TOTAL  (WS_SIG + 96 * 8)   // = 1088 floats

// ---- wave-level sum on the matrix pipe ------------------------------------
// A (16x4 f32 layout): lane L<16 holds A[L][0],A[L][1]; lane L>=16 holds
// A[L-16][2],A[L-16][3]. With a={v,0}, b=ones: D[m][n] = v_m + v_{m+16}.
// Summing the 8 D VGPRs gives half-sums per lane-half; xor-16 completes it.
__device__ __forceinline__ float wave_sum_wmma(float v) {
  v2f a; a.x = v;    a.y = 0.0f;
  v2f b; b.x = 1.0f; b.y = 1.0f;
  v8f c = {0.0f, 0.0f, 0.0f, 0.0f, 0.0f, 0.0f, 0.0f, 0.0f};
  c = __builtin_amdgcn_wmma_f32_16x16x4_f32(false, a, false, b,
                                            (short)0, c, false, false);
  float s = c[0] + c[1] + c[2] + c[3] + c[4] + c[5] + c[6] + c[7];
  s += __shfl_xor(s, 16, 32);
  return s;
}

__device__ __forceinline__ float wave_min32(float v) {
#pragma unroll
  for (int o = 16; o > 0; o >>= 1) v = fminf(v, __shfl_xor(v, o, 32));
  return v;
}
__device__ __forceinline__ float wave_max32(float v) {
#pragma unroll
  for (int o = 16; o > 0; o >>= 1) v = fmaxf(v, __shfl_xor(v, o, 32));
  return v;
}

// block reductions (blockDim.x == 256 => 8 waves); result broadcast to all.
__device__ float block_sum(float v) {
  __shared__ float tmp[8];
  const int lane = threadIdx.x & 31, wid = threadIdx.x >> 5;
  v = wave_sum_wmma(v);
  __syncthreads();
  if (lane == 0) tmp[wid] = v;
  __syncthreads();
  if (wid == 0) {
    float t = (lane < (int)(blockDim.x >> 5)) ? tmp[lane] : 0.0f;
    t = wave_sum_wmma(t);
    if (lane == 0) tmp[0] = t;
  }
  __syncthreads();
  return tmp[0];
}

__device__ float block_min(float v) {
  __shared__ float tmpm[8];
  const int lane = threadIdx.x & 31, wid = threadIdx.x >> 5;
  v = wave_min32(v);
  __syncthreads();
  if (lane == 0) tmpm[wid] = v;
  __syncthreads();
  if (wid == 0) {
    float t = (lane < (int)(blockDim.x >> 5)) ? tmpm[lane] : __builtin_huge_valf();
    t = wave_min32(t);
    if (lane == 0) tmpm[0] = t;
  }
  __syncthreads();
  return tmpm[0];
}

__device__ float block_max(float v) {
  __shared__ float tmpx[8];
  const int lane = threadIdx.x & 31, wid = threadIdx.x >> 5;
  v = wave_max32(v);
  __syncthreads();
  if (lane == 0) tmpx[wid] = v;
  __syncthreads();
  if (wid == 0) {
    float t = (lane < (int)(blockDim.x >> 5)) ? tmpx[lane] : -__builtin_huge_valf();
    t = wave_max32(t);
    if (lane == 0) tmpx[0] = t;
  }
  __syncthreads();
  return tmpx[0];
}

// ---- kernels ---------------------------------------------------------------

__global__ void zero_ws_kernel(float* ws, int n) {
  for (int i = blockIdx.x * blockDim.x + threadIdx.x; i < n;
       i += gridDim.x * blockDim.x)
    ws[i] = 0.0f;
}

// seg: fused CE(ignore=255) + softmax statistics for Dice, one pass over the
// 109MB logit tensor. blockIdx.y = batch so per-(b,c) sums reduce per block.
__global__ void seg_kernel(const float* __restrict__ logits,
                           const int* __restrict__ tgt,
                           float* __restrict__ ws) {
  const int b = blockIdx.y;
  const float* base = logits + (size_t)b * NC * HW;
  const int* tb = tgt + (size_t)b * HW;

  float sump[NC], inter[NC], cnt[NC];
#pragma unroll
  for (int c = 0; c < NC; c++) { sump[c] = 0.0f; inter[c] = 0.0f; cnt[c] = 0.0f; }
  float ce_s = 0.0f, ce_c = 0.0f;

  for (int p = blockIdx.x * blockDim.x + threadIdx.x; p < HW;
       p += gridDim.x * blockDim.x) {
    float x[NC];
#pragma unroll
    for (int c = 0; c < NC; c++) x[c] = base[(size_t)c * HW + p];
    float mx = x[0];
#pragma unroll
    for (int c = 1; c < NC; c++) mx = fmaxf(mx, x[c]);
    float e[NC];
    float s = 0.0f;
#pragma unroll
    for (int c = 0; c < NC; c++) { e[c] = __expf(x[c] - mx); s += e[c]; }
    const float rs = 1.0f / s;
    const int t = tb[p];
    float xt = 0.0f;
#pragma unroll
    for (int c = 0; c < NC; c++) {
      const float pc = e[c] * rs;
      sump[c] += pc;
      const bool is = (c == t);
      inter[c] += is ? pc : 0.0f;     // one_hot(255) contributes nothing
      cnt[c]   += is ? 1.0f : 0.0f;
      if (is) xt = x[c];              // avoids dynamic reg indexing / spills
    }
    if (t != 255) { ce_s += (mx + __logf(s)) - xt; ce_c += 1.0f; }
  }

  ce_s = block_sum(ce_s);
  ce_c = block_sum(ce_c);
  if (threadIdx.x == 0) {
    atomicAdd(&ws[WS_CE_SUM], ce_s);
    atomicAdd(&ws[WS_CE_CNT], ce_c);
  }
  for (int c = 0; c < NC; c++) {
    const float si = block_sum(inter[c]);
    const float sp = block_sum(sump[c]);
    const float sc = block_sum(cnt[c]);
    if (threadIdx.x == 0) {
      atomicAdd(&ws[WS_INTER + b * NC + c], si);
      atomicAdd(&ws[WS_SUMP  + b * NC + c], sp);
      atomicAdd(&ws[WS_CNTOH + b * NC + c], sc);
    }
  }
}

// grid: BCE + global dice sums, single streaming pass.
__global__ void grid_kernel(const float* __restrict__ pr,
                            const float* __restrict__ tr,
                            float* __restrict__ ws, int n) {
  float bce = 0.0f, inter = 0.0f, sp = 0.0f, st = 0.0f;
  for (int i = blockIdx.x * blockDim.x + threadIdx.x; i < n;
       i += gridDim.x * blockDim.x) {
    const float p = pr[i], t = tr[i];
    const float lp  = fmaxf(__logf(fminf(fmaxf(p, 1e-12f), 1.0f)), -100.0f);
    const float l1p = fmaxf(__logf(fminf(fmaxf(1.0f - p, 1e-12f), 1.0f)), -100.0f);
    bce  += -(t * lp + (1.0f - t) * l1p);
    inter += p * t;
    sp += p;
    st += t;
  }
  bce = block_sum(bce);
  inter = block_sum(inter);
  sp = block_sum(sp);
  st = block_sum(st);
  if (threadIdx.x == 0) {
    atomicAdd(&ws[WS_GRID + 0], bce);
    atomicAdd(&ws[WS_GRID + 1], inter);
    atomicAdd(&ws[WS_GRID + 2], sp);
    atomicAdd(&ws[WS_GRID + 3], st);
  }
}

// baseline: gather only the 4 bilinear taps per 32x32 output (align_corners),
// so the 100MB target tensor is NOT streamed.
__global__ void baseline_kernel(const float* __restrict__ pred,
                                const float* __restrict__ tgt,
                                float* __restrict__ ws) {
  const int NOUT = NBATCH * NL * 32 * 32;
  const float step = 511.0f / 31.0f;
  float bce = 0.0f;
  for (int e = blockIdx.x * blockDim.x + threadIdx.x; e < NOUT;
       e += gridDim.x * blockDim.x) {
    const int ox = e & 31, oy = (e >> 5) & 31;
    const int bl = e >> 10;  // b*12 + l
    const float fy = oy * step, fx = ox * step;
    const int y0 = (int)floorf(fy), x0 = (int)floorf(fx);
    const int y1 = min(y0 + 1, 511), x1 = min(x0 + 1, 511);
    const float wy = fy - (float)y0, wx = fx - (float)x0;
    const float* tb = tgt + (size_t)bl * HW;
    const float t00 = tb[y0 * 512 + x0], t01 = tb[y0 * 512 + x1];
    const float t10 = tb[y1 * 512 + x0], t11 = tb[y1 * 512 + x1];
    const float r0 = t00 * (1.0f - wy) + t10 * wy;
    const float r1 = t01 * (1.0f - wy) + t11 * wy;
    const float t = r0 * (1.0f - wx) + r1 * wx;
    const float p = pred[e];
    const float lp  = fmaxf(__logf(fminf(fmaxf(p, 1e-12f), 1.0f)), -100.0f);
    const float l1p = fmaxf(__logf(fminf(fmaxf(1.0f - p, 1e-12f), 1.0f)), -100.0f);
    bce += -(t * lp + (1.0f - t) * l1p);
  }
  bce = block_sum(bce);
  if (threadIdx.x == 0) atomicAdd(&ws[WS_BASE], bce);
}

// signal: one block per (b,lead). Pass1: masked min/max/count. Pass2: moment
// sums (Pearson expanded algebraically so no third pass for means).
__global__ void signal_kernel(const float* __restrict__ pred,
                              const float* __restrict__ gt,
                              const unsigned char* __restrict__ mask,
                              float* __restrict__ ws) {
  const int pair = blockIdx.x;            // 0..95
  const int b = pair / NL, l = pair % NL;
  const float* pp = pred + ((size_t)b * NL + l) * TSIG;          // (B,12,T)
  const float* gp = gt + (size_t)b * TSIG * NL + l;              // (B,T,12)
  const unsigned char* mp = mask + (size_t)b * TSIG * NL + l;    // (B,T,12)

  float cnt = 0.0f;
  float mnp = __builtin_huge_valf(), mxp = -__builtin_huge_valf();
  float mnt = __builtin_huge_valf(), mxt = -__builtin_huge_valf();
  for (int t = threadIdx.x; t < TSIG; t += blockDim.x) {
    if (mp[(size_t)t * NL]) {
      const float p = pp[t], g = gp[(size_t)t * NL];
      cnt += 1.0f;
      mnp = fminf(mnp, p); mxp = fmaxf(mxp, p);
      mnt = fminf(mnt, g); mxt = fmaxf(mxt, g);
    }
  }
  cnt = block_sum(cnt);
  mnp = block_min(mnp); mxp = block_max(mxp);
  mnt = block_min(mnt); mxt = block_max(mxt);

  const float rngp = mxp - mnp, rngt = mxt - mnt;
  const bool okp = (cnt >= 2.0f) && (rngp >= 1e-6f);
  const bool okt = (cnt >= 2.0f) && (rngt >= 1e-6f);
  const float mnp_s = okp ? mnp : 0.0f, rp = okp ? rngp : 1.0f;
  const float mnt_s = okt ? mnt : 0.0f, rt = okt ? rngt : 1.0f;

  float sabs = 0, sp = 0, st = 0, spt = 0, sp2 = 0, st2 = 0;
  for (int t = threadIdx.x; t < TSIG; t += blockDim.x) {
    if (mp[(size_t)t * NL]) {
      const float p = pp[t], g = gp[(size_t)t * NL];
      const float pn = okp ? (2.0f * (p - mnp_s) / (rp + 1e-8f) - 1.0f) : 0.0f;
      const float tn = okt ? (2.0f * (g - mnt_s) / (rt + 1e-8f) - 1.0f) : 0.0f;
      sabs += fabsf(pn - tn);
      sp += pn; st += tn;
      spt += pn * tn; sp2 += pn * pn; st2 += tn * tn;
    }
  }
  sabs = block_sum(sabs);
  sp = block_sum(sp);   st = block_sum(st);
  spt = block_sum(spt); sp2 = block_sum(sp2); st2 = block_sum(st2);
  if (threadIdx.x == 0) {
    float* o = ws + WS_SIG + pair * 8;
    o[0] = sabs; o[1] = cnt; o[2] = sp; o[3] = st;
    o[4] = spt;  o[5] = sp2; o[6] = st2;
  }
}

// finalize: a few hundred scalar ops; single thread for exact sequencing.
__global__ void final_kernel(const float* __restrict__ ws,
                             const float* __restrict__ tp,
                             const float* __restrict__ tg,
                             float* __restrict__ out) {
  const float ce = ws[WS_CE_SUM] / fmaxf(ws[WS_CE_CNT], 1.0f);
  float dsum = 0.0f;
  for (int i = 0; i < 104; i++) {
    const float inter = ws[WS_INTER + i];
    const float un = ws[WS_SUMP + i] + ws[WS_CNTOH + i];
    dsum += (2.0f * inter + 1e-7f) / (un + 1e-7f);
  }
  const float seg = (ce + (1.0f - dsum / 104.0f)) * 1.0f;

  const float gb = ws[WS_GRID + 0] / (float)(NBATCH * HW);
  const float gd = 1.0f - (2.0f * ws[WS_GRID + 1] + 1e-7f) /
                              (ws[WS_GRID + 2] + ws[WS_GRID + 3] + 1e-7f);
  const float grid = (gb + gd) * 0.5f;

  const float baseline = (ws[WS_BASE] / (float)(NBATCH * NL * 32 * 32)) * 0.8f;

  float th = 0.0f;
  for (int i = 0; i < 48; i++) th += fabsf(tp[i] - tg[i]);
  const float theta = (th / 48.0f) * 0.3f;

  float sabs = 0.0f, sm = 0.0f, csum = 0.0f, nv = 0.0f;
  for (int p = 0; p < 96; p++) {
    const float* o = ws + WS_SIG + p * 8;
    const float a = o[0], cnt = o[1], spn = o[2], stn = o[3];
    const float spt = o[4], sp2 = o[5], st2 = o[6];
    sabs += a; sm += cnt;
    const float cs = fmaxf(cnt, 1.0f);
    const float pm = spn / cs, tm = stn / cs;
    const float num = spt - cnt * pm * tm;
    const float p2 = fmaxf(sp2 - cnt * pm * pm, 0.0f);
    const float t2 = fmaxf(st2 - cnt * tm * tm, 0.0f);
    const float ps = sqrtf(p2 + 1e-6f), ts = sqrtf(t2 + 1e-6f);
    const float corr = fminf(fmaxf(num / (ps * ts + 1e-6f), -1.0f), 1.0f);
    if (cnt >= 10.0f) { csum += corr; nv += 1.0f; }
  }
  const float mae = sabs / (sm + 1e-7f);
  const float avg = csum / fmaxf(nv, 1.0f);
  const float pear = (nv > 0.0f) ? (1.0f - avg) : 1.0f;
  const float signal = (mae + pear) * 2.0f;

  out[0] = seg; out[1] = grid; out[2] = baseline;
  out[3] = theta; out[4] = signal;
  out[5] = seg + grid + baseline + theta + signal;
}

// ---------------------------------------------------------------------------
extern "C" void kernel_launch(void* const* d_in, const int* in_sizes, int n_in,
                              void* d_out, int out_size, void* d_ws, size_t ws_size,
                              hipStream_t stream) {
  const float* seg_logits = (const float*)d_in[0];
  const int*   seg_tgt    = (const int*)d_in[1];
  const float* grid_p     = (const float*)d_in[2];
  const float* grid_t     = (const float*)d_in[3];
  const float* base_p     = (const float*)d_in[4];
  const float* base_t     = (const float*)d_in[5];
  const float* theta_p    = (const float*)d_in[6];
  const float* theta_g    = (const float*)d_in[7];
  const float* sig_p      = (const float*)d_in[8];
  const float* sig_g      = (const float*)d_in[9];
  const unsigned char* sig_m = (const unsigned char*)d_in[10];  // numpy bool (1B)
  float* out = (float*)d_out;
  float* ws  = (float*)d_ws;

  zero_ws_kernel<<<4, 256, 0, stream>>>(ws, WS_TOTAL);
  dim3 sg(128, NBATCH);
  seg_kernel<<<sg, 256, 0, stream>>>(seg_logits, seg_tgt, ws);
  grid_kernel<<<512, 256, 0, stream>>>(grid_p, grid_t, ws, NBATCH * HW);
  baseline_kernel<<<96, 256, 0, stream>>>(base_p, base_t, ws);
  signal_kernel<<<96, 256, 0, stream>>>(sig_p, sig_g, sig_m, ws);
  final_kernel<<<1, 1, 0, stream>>>(ws, theta_p, theta_g, out);
}